// Net_46316927320323
// MI455X (gfx1250) — compile-verified
//
#include <hip/hip_runtime.h>
#include <math.h>

#define N_NODES 100000
#define N_FEAT  100
#define HID     32
#define N_CLS   2
#define N_EDGES 1600000
#define EA      (N_EDGES + N_NODES)   // edges + self loops

typedef __attribute__((ext_vector_type(2))) float v2f;
typedef __attribute__((ext_vector_type(8))) float v8f;

__device__ __forceinline__ float wave_sum(float v) {
#pragma unroll
    for (int off = 16; off > 0; off >>= 1) v += __shfl_xor(v, off, 32);
    return v;
}

// -------------------------------------------------------------------------
// K1: x = relu(X @ W1^T + b1)  via V_WMMA_F32_16X16X4_F32
// One wave -> 16 rows x 32 cols (two 16x16 C tiles), 25 k-steps of K=4.
// A (16x4 f32): lanes 0-15 hold K={k,k+1} of row lane; lanes 16-31 K={k+2,k+3}.
// B (4x16 f32): mirrored; B(k,n) = W1[n*N_FEAT + k].
// C/D (16x16 f32, 8 VGPRs): VGPR v -> M = v + 8*(lane>>4), N = lane&15.
// -------------------------------------------------------------------------
__global__ void lin1_wmma_kernel(const float* __restrict__ X,
                                 const float* __restrict__ W1,
                                 const float* __restrict__ B1,
                                 float* __restrict__ xout) {
    const int lane  = threadIdx.x & 31;
    const int wid   = threadIdx.x >> 5;
    const int tile  = blockIdx.x * (blockDim.x >> 5) + wid;
    const int ntile = N_NODES / 16;              // 6250 exactly
    if (tile >= ntile) return;                   // wave-uniform: EXEC stays all-1s

    const int r0    = tile * 16;
    const int mrow  = r0 + (lane & 15);
    const int khalf = lane >> 4;                 // 0 -> K={k,k+1}, 1 -> K={k+2,k+3}
    const int ncol  = lane & 15;

    const float* xrow = X  + (size_t)mrow * N_FEAT;
    const float* wlo  = W1 + (size_t)ncol * N_FEAT;          // cols 0..15
    const float* whi  = W1 + (size_t)(ncol + 16) * N_FEAT;   // cols 16..31

    v8f c0 = {};
    v8f c1 = {};
#pragma unroll 5
    for (int k = 0; k < N_FEAT; k += 4) {
        const int kk = k + khalf * 2;            // 8-byte aligned (kk even)
        v2f a  = *(const v2f*)(xrow + kk);
        v2f b0 = *(const v2f*)(wlo  + kk);
        v2f b1 = *(const v2f*)(whi  + kk);
        c0 = __builtin_amdgcn_wmma_f32_16x16x4_f32(false, a, false, b0,
                                                   (short)0, c0, false, false);
        c1 = __builtin_amdgcn_wmma_f32_16x16x4_f32(false, a, false, b1,
                                                   (short)0, c1, false, false);
    }

    const float bias0 = B1[ncol];
    const float bias1 = B1[ncol + 16];
#pragma unroll
    for (int v = 0; v < 8; ++v) {
        const int orow = r0 + v + 8 * khalf;
        float t0 = c0[v] + bias0; t0 = t0 > 0.f ? t0 : 0.f;
        float t1 = c1[v] + bias1; t1 = t1 > 0.f ? t1 : 0.f;
        xout[(size_t)orow * HID + ncol]      = t0;
        xout[(size_t)orow * HID + ncol + 16] = t1;
    }
}

// -------------------------------------------------------------------------
// K2: xn = x / max(||x||, 1e-12)   (one wave per row, coalesced)
// -------------------------------------------------------------------------
__global__ void norm_kernel(const float* __restrict__ x, float* __restrict__ xn) {
    const int lane = threadIdx.x & 31;
    const int row  = blockIdx.x * (blockDim.x >> 5) + (threadIdx.x >> 5);
    if (row >= N_NODES) return;
    const float v = x[(size_t)row * HID + lane];
    const float s = wave_sum(v * v);
    const float inv = 1.0f / fmaxf(sqrtf(s), 1e-12f);
    xn[(size_t)row * HID + lane] = v * inv;
}

// -------------------------------------------------------------------------
// K3: per-edge cosine attention. alpha in [-beta, beta] so softmax-max
// subtraction is a mathematical no-op -> compute exp(alpha) directly,
// accumulate denom[tgt]. One wave per edge; gathers are L2-resident.
// -------------------------------------------------------------------------
__global__ void edge_alpha_kernel(const float* __restrict__ xn,
                                  const int* __restrict__ ei,
                                  const float* __restrict__ beta_p,
                                  float* __restrict__ expw,
                                  float* __restrict__ denom) {
    const int lane = threadIdx.x & 31;
    const int e    = blockIdx.x * (blockDim.x >> 5) + (threadIdx.x >> 5);
    if (e >= EA) return;
    int src, tgt;
    if (e < N_EDGES) { src = ei[e]; tgt = ei[N_EDGES + e]; }
    else             { src = e - N_EDGES; tgt = src; }      // self loop
    const float p   = xn[(size_t)tgt * HID + lane] * xn[(size_t)src * HID + lane];
    const float dot = wave_sum(p);
    if (lane == 0) {
        const float ew = expf(beta_p[0] * dot);
        expw[e] = ew;
        atomicAdd(&denom[tgt], ew);                         // global_atomic_add_f32
    }
}

// -------------------------------------------------------------------------
// K4: out[tgt] += (expw[e]/denom[tgt]) * x[src]  (one wave per edge,
// 32 lanes = 32 features; f32 atomics into L2-resident 12.8MB buffer)
// -------------------------------------------------------------------------
__global__ void aggregate_kernel(const float* __restrict__ x,
                                 const int* __restrict__ ei,
                                 const float* __restrict__ expw,
                                 const float* __restrict__ denom,
                                 float* __restrict__ outf) {
    const int lane = threadIdx.x & 31;
    const int e    = blockIdx.x * (blockDim.x >> 5) + (threadIdx.x >> 5);
    if (e >= EA) return;
    int src, tgt;
    if (e < N_EDGES) { src = ei[e]; tgt = ei[N_EDGES + e]; }
    else             { src = e - N_EDGES; tgt = src; }
    const float w = expw[e] / denom[tgt];
    atomicAdd(&outf[(size_t)tgt * HID + lane], w * x[(size_t)src * HID + lane]);
}

// -------------------------------------------------------------------------
// K5: logits = outf @ W2^T + b2 ; log_softmax over 2 classes.
// One wave per node, two width-32 shuffle reductions.
// -------------------------------------------------------------------------
__global__ void head_kernel(const float* __restrict__ outf,
                            const float* __restrict__ W2,
                            const float* __restrict__ B2,
                            float* __restrict__ out) {
    const int lane = threadIdx.x & 31;
    const int row  = blockIdx.x * (blockDim.x >> 5) + (threadIdx.x >> 5);
    if (row >= N_NODES) return;
    const float v  = outf[(size_t)row * HID + lane];
    const float d0 = wave_sum(v * W2[lane]);
    const float d1 = wave_sum(v * W2[HID + lane]);
    if (lane == 0) {
        const float l0 = d0 + B2[0];
        const float l1 = d1 + B2[1];
        const float m  = fmaxf(l0, l1);
        const float lse = m + logf(expf(l0 - m) + expf(l1 - m));
        out[(size_t)row * N_CLS + 0] = l0 - lse;
        out[(size_t)row * N_CLS + 1] = l1 - lse;
    }
}

extern "C" void kernel_launch(void* const* d_in, const int* in_sizes, int n_in,
                              void* d_out, int out_size, void* d_ws, size_t ws_size,
                              hipStream_t stream) {
    const float* X    = (const float*)d_in[0];   // [N, 100]
    const float* W1   = (const float*)d_in[1];   // [32, 100]
    const float* B1   = (const float*)d_in[2];   // [32]
    const float* beta = (const float*)d_in[3];   // [1]
    const float* W2   = (const float*)d_in[4];   // [2, 32]
    const float* B2   = (const float*)d_in[5];   // [2]
    const int*   ei   = (const int*)d_in[6];     // [2, E] flat: src then tgt
    float*       out  = (float*)d_out;           // [N, 2]

    // workspace carve-up (~46 MB of f32)
    float* ws    = (float*)d_ws;
    float* x     = ws;                               // N*32
    float* xn    = x    + (size_t)N_NODES * HID;     // N*32
    float* expw  = xn   + (size_t)N_NODES * HID;     // E+N
    float* denom = expw + (size_t)EA;                // N
    float* outf  = denom + (size_t)N_NODES;          // N*32

    hipMemsetAsync(denom, 0, (size_t)N_NODES * sizeof(float), stream);
    hipMemsetAsync(outf,  0, (size_t)N_NODES * HID * sizeof(float), stream);

    {   // lin1 + relu (WMMA): 6250 wave-tiles, 4 waves/block
        const int tiles = N_NODES / 16;
        const int wpb = 4;
        lin1_wmma_kernel<<<dim3((tiles + wpb - 1) / wpb), dim3(128), 0, stream>>>(
            X, W1, B1, x);
    }
    {   // row normalize: 1 wave/row, 8 waves/block
        const int wpb = 8;
        norm_kernel<<<dim3((N_NODES + wpb - 1) / wpb), dim3(256), 0, stream>>>(x, xn);
    }
    {   // edge passes: 1 wave/edge, 8 waves/block
        const int wpb = 8;
        const dim3 grd((EA + wpb - 1) / wpb), blk(256);
        edge_alpha_kernel<<<grd, blk, 0, stream>>>(xn, ei, beta, expw, denom);
        aggregate_kernel<<<grd, blk, 0, stream>>>(x, ei, expw, denom, outf);
    }
    {   // head + log_softmax: 1 wave/row
        const int wpb = 8;
        head_kernel<<<dim3((N_NODES + wpb - 1) / wpb), dim3(256), 0, stream>>>(
            outf, W2, B2, out);
    }
}